// audio_encoder_24395414241964
// MI455X (gfx1250) — compile-verified
//
#include <hip/hip_runtime.h>
#include <math.h>

// MI455X / gfx1250, wave32. fp32 WMMA (16x16x4) implicit-GEMM TDNN stack +
// sequential GRU scans. Workload is HBM-bandwidth bound (~2GB traffic vs
// 384 GFLOP), so we stay in fp32 on the matrix pipe — no precision loss.
// Staging uses GLOBAL_LOAD_ASYNC_TO_LDS_B128 (CDNA5 async copy, ASYNCcnt)
// on the fast path, scalar gather on tails / layer-1's Din=20.

typedef __attribute__((ext_vector_type(2))) float v2f;
typedef __attribute__((ext_vector_type(8))) float v8f;

#if __has_builtin(__builtin_amdgcn_global_load_async_to_lds_b128)
#define HAVE_ASYNC_LDS 1
typedef int v4i __attribute__((vector_size(16)));
typedef __attribute__((address_space(1))) v4i* g_v4i_p;   // global ("__device__")
typedef __attribute__((address_space(3))) v4i* l_v4i_p;   // LDS ("__shared__")
#else
#define HAVE_ASYNC_LDS 0
#endif

#define BN_EPS 1e-5f

// ---------------- transpose (B,C,L) -> (B,L,C) ----------------
__global__ __launch_bounds__(256) void transpose_bcl_blc(
    const float* __restrict__ x, float* __restrict__ y, int Bn, int C, int L)
{
    long long n = (long long)Bn * C * L;
    for (long long i = blockIdx.x * (long long)blockDim.x + threadIdx.x; i < n;
         i += (long long)gridDim.x * blockDim.x) {
        int l = (int)(i % L);
        long long t = i / L;
        int c = (int)(t % C);
        int b = (int)(t / C);
        y[((long long)b * L + l) * C + c] = x[i];
    }
}

// ---------------- implicit-GEMM TDNN layer (fp32 WMMA) ----------------
// out[b, l, o] = BN(ReLU( sum_{tap,k} in[b, l + tap*dil, k] * w[o, tap*Din+k] + bias[o] ))
// mode 0: bias + ReLU + BN(eval).  mode 1: bias only (plain GEMM).
#define MT 64
#define NT 128
#define KB 32
#define KPAD 36   // +4 float pad: rows 16B-aligned for b128 async copies, and
                  // column reads hit distinct banks (gcd(36,64)=4 -> 16 banks,
                  // half-wave +2-dword offset uses the disjoint =2 mod 4 set)

__global__ __launch_bounds__(256) void tdnn_wmma_f32(
    const float* __restrict__ in,    // (Bn, Lin, Din)
    const float* __restrict__ w,     // (Dout, K = c*Din)
    const float* __restrict__ bias,  // (Dout)
    const float* __restrict__ g, const float* __restrict__ bb,
    const float* __restrict__ mean, const float* __restrict__ var,
    float* __restrict__ out,         // (Bn, Lout, Dout)
    int Bn, int Lin, int Lout, int Din, int Dout, int c, int dil, int mode)
{
    const int K = c * Din;
    const int M = Bn * Lout;
    const int m0 = blockIdx.x * MT;
    const int n0 = blockIdx.y * NT;

    __shared__ float As[MT * KPAD];
    __shared__ float Bs[NT * KPAD];

    const int tid  = threadIdx.x;
    const int lane = tid & 31;
    const int wave = tid >> 5;
    const int wm   = wave & 3;          // wave's 16-row M subtile (4 along M)
    const int wn   = wave >> 2;         // wave's 64-col N subtile (2 along N)
    const int lm   = lane & 15;
    const int koff = (lane >> 4) * 2;   // lanes 16..31 hold K+2 / K+3

    // A fast path: K-block contiguous (no tap crossing) and full M tile.
    const bool a_async_ok = ((Din & (KB - 1)) == 0) && (m0 + MT <= M);

    v8f zero = {0.f, 0.f, 0.f, 0.f, 0.f, 0.f, 0.f, 0.f};
    v8f acc[4];
    acc[0] = zero; acc[1] = zero; acc[2] = zero; acc[3] = zero;

    for (int k0 = 0; k0 < K; k0 += KB) {
        // ---- stage A tile (dilated taps), 64x32 ----
#if HAVE_ASYNC_LDS
        if (a_async_ok && (k0 + KB <= K)) {
            const int tap = k0 / Din;
            const int kk0 = k0 - tap * Din;
            for (int e = tid; e < MT * (KB / 4); e += 256) {   // 512 b128 chunks
                int mm = e >> 3;
                int j  = (e & 7) * 4;
                int mg = m0 + mm;
                int b  = mg / Lout;
                int l  = mg - b * Lout;
                const float* src =
                    in + ((long long)b * Lin + (l + tap * dil)) * Din + kk0 + j;
                __builtin_amdgcn_global_load_async_to_lds_b128(
                    (g_v4i_p)(void*)src, (l_v4i_p)(void*)&As[mm * KPAD + j], 0, 0);
            }
        } else
#endif
        {
            for (int e = tid; e < MT * KB; e += 256) {
                int mm = e >> 5;            // row in tile (KB == 32)
                int kk = e & (KB - 1);
                int kg = k0 + kk;
                int mg = m0 + mm;
                float val = 0.f;
                if (mg < M && kg < K) {
                    int tap = kg / Din;
                    int fch = kg - tap * Din;
                    int b   = mg / Lout;
                    int l   = mg - b * Lout;
                    val = in[((long long)b * Lin + (l + tap * dil)) * Din + fch];
                }
                As[mm * KPAD + kk] = val;
            }
        }
        // ---- stage B tile (weights, contiguous rows over K), 128x32 ----
#if HAVE_ASYNC_LDS
        if ((n0 + NT <= Dout) && (k0 + KB <= K)) {
            for (int e = tid; e < NT * (KB / 4); e += 256) {   // 1024 b128 chunks
                int nn = e >> 3;
                int j  = (e & 7) * 4;
                const float* src = w + (long long)(n0 + nn) * K + k0 + j;
                __builtin_amdgcn_global_load_async_to_lds_b128(
                    (g_v4i_p)(void*)src, (l_v4i_p)(void*)&Bs[nn * KPAD + j], 0, 0);
            }
        } else
#endif
        {
            for (int e = tid; e < NT * KB; e += 256) {
                int nn = e >> 5;
                int kk = e & (KB - 1);
                int kg = k0 + kk;
                int ng = n0 + nn;
                float val = (ng < Dout && kg < K) ? w[(long long)ng * K + kg] : 0.f;
                Bs[nn * KPAD + kk] = val;
            }
        }
#if HAVE_ASYNC_LDS
#if __has_builtin(__builtin_amdgcn_s_wait_asynccnt)
        __builtin_amdgcn_s_wait_asynccnt(0);
#else
        asm volatile("s_wait_asynccnt 0x0" ::: "memory");
#endif
#endif
        __syncthreads();

        const float* arow = &As[(wm * 16 + lm) * KPAD];
#pragma unroll
        for (int kk = 0; kk < KB; kk += 4) {
            v2f a;                       // A 16x4 f32 fragment (2 VGPRs/lane)
            a.x = arow[kk + koff];
            a.y = arow[kk + koff + 1];
#pragma unroll
            for (int j = 0; j < 4; ++j) {
                const float* brow = &Bs[(wn * 64 + j * 16 + lm) * KPAD];
                v2f bf;                  // B 4x16 f32 fragment
                bf.x = brow[kk + koff];
                bf.y = brow[kk + koff + 1];
                acc[j] = __builtin_amdgcn_wmma_f32_16x16x4_f32(
                    false, a, false, bf, (short)0, acc[j], false, false);
            }
        }
        __syncthreads();
    }

    // ---- fused epilogue: bias (+ ReLU + BN) ----
    const int mbase = m0 + wm * 16 + (lane >> 4) * 8;  // C: M split 0-7 / 8-15
#pragma unroll
    for (int j = 0; j < 4; ++j) {
        int ng = n0 + wn * 64 + j * 16 + lm;
        if (ng >= Dout) continue;
        float bv = bias[ng];
        float sc = 1.f, sh = 0.f;
        if (mode == 0) {
            float s = g[ng] * rsqrtf(var[ng] + BN_EPS);
            sc = s;
            sh = bb[ng] - mean[ng] * s;
        }
#pragma unroll
        for (int r = 0; r < 8; ++r) {
            int mg = mbase + r;
            if (mg < M) {
                float y = acc[j][r] + bv;
                if (mode == 0) { y = fmaxf(y, 0.f); y = y * sc + sh; }
                out[(long long)mg * Dout + ng] = y;
            }
        }
    }
}

// ---------------- GRU recurrent scans (H=16, B=32) ----------------
__device__ __forceinline__ float sigm(float x) { return 1.f / (1.f + expf(-x)); }

// Layer 0: input projections gi precomputed (B,T,48). One thread per (b,h).
__global__ __launch_bounds__(512) void gru_scan0(
    const float* __restrict__ gi, const float* __restrict__ whh,
    const float* __restrict__ bhh, float* __restrict__ ys, int T)
{
    __shared__ float h[32][16];
    __shared__ float wl[48][16];
    __shared__ float bl[48];
    const int t = threadIdx.x;
    const int b = t >> 4, hi = t & 15;
    for (int e = t; e < 48 * 16; e += 512) wl[e >> 4][e & 15] = whh[e];
    if (t < 48) bl[t] = bhh[t];
    h[b][hi] = 0.f;
    __syncthreads();
    for (int step = 0; step < T; ++step) {
        float hr = 0.f, hz = 0.f, hn = 0.f;
#pragma unroll
        for (int k = 0; k < 16; ++k) {
            float hv = h[b][k];
            hr += wl[hi][k] * hv;
            hz += wl[16 + hi][k] * hv;
            hn += wl[32 + hi][k] * hv;
        }
        long long base = ((long long)b * T + step) * 48;
        float r = sigm(gi[base + hi]      + hr + bl[hi]);
        float z = sigm(gi[base + 16 + hi] + hz + bl[16 + hi]);
        float n = tanhf(gi[base + 32 + hi] + r * (hn + bl[32 + hi]));
        float hnew = (1.f - z) * n + z * h[b][hi];
        __syncthreads();
        h[b][hi] = hnew;
        ys[((long long)b * T + step) * 16 + hi] = hnew;
        __syncthreads();
    }
}

// Layer 1: input dim 16, compute input projection inline per step.
__global__ __launch_bounds__(512) void gru_scan1(
    const float* __restrict__ x, const float* __restrict__ wih,
    const float* __restrict__ whh, const float* __restrict__ bih,
    const float* __restrict__ bhh, float* __restrict__ out, int T)
{
    __shared__ float h[32][16];
    __shared__ float xt[32][16];
    __shared__ float wi[48][16], wl[48][16];
    __shared__ float bi[48], bl[48];
    const int t = threadIdx.x;
    const int b = t >> 4, hi = t & 15;
    for (int e = t; e < 48 * 16; e += 512) {
        wi[e >> 4][e & 15] = wih[e];
        wl[e >> 4][e & 15] = whh[e];
    }
    if (t < 48) { bi[t] = bih[t]; bl[t] = bhh[t]; }
    h[b][hi] = 0.f;
    __syncthreads();
    for (int step = 0; step < T; ++step) {
        xt[b][hi] = x[((long long)b * T + step) * 16 + hi];
        __syncthreads();
        float ir = 0.f, iz = 0.f, in_ = 0.f, hr = 0.f, hz = 0.f, hn = 0.f;
#pragma unroll
        for (int k = 0; k < 16; ++k) {
            float xv = xt[b][k], hv = h[b][k];
            ir  += wi[hi][k] * xv;       hr += wl[hi][k] * hv;
            iz  += wi[16 + hi][k] * xv;  hz += wl[16 + hi][k] * hv;
            in_ += wi[32 + hi][k] * xv;  hn += wl[32 + hi][k] * hv;
        }
        float r = sigm(ir + bi[hi]      + hr + bl[hi]);
        float z = sigm(iz + bi[16 + hi] + hz + bl[16 + hi]);
        float n = tanhf(in_ + bi[32 + hi] + r * (hn + bl[32 + hi]));
        float hnew = (1.f - z) * n + z * h[b][hi];
        __syncthreads();
        h[b][hi] = hnew;
        out[((long long)b * T + step) * 16 + hi] = hnew;  // (B, T*H) flat
        __syncthreads();
    }
}

// ---------------- host orchestration ----------------
extern "C" void kernel_launch(void* const* d_in, const int* in_sizes, int n_in,
                              void* d_out, int out_size, void* d_ws, size_t ws_size,
                              hipStream_t stream)
{
    (void)in_sizes; (void)n_in; (void)out_size; (void)ws_size;
    const int Bn = 32, L = 4096, T = 4024;

    const float* x = (const float*)d_in[0];
    const float *W[5], *Bi[5], *G[5], *BB[5], *Mn[5], *Vr[5];
    for (int i = 0; i < 5; ++i) {
        int base = 1 + i * 6;
        W[i]  = (const float*)d_in[base + 0];
        Bi[i] = (const float*)d_in[base + 1];
        G[i]  = (const float*)d_in[base + 2];
        BB[i] = (const float*)d_in[base + 3];
        Mn[i] = (const float*)d_in[base + 4];
        Vr[i] = (const float*)d_in[base + 5];
    }
    const float* wih0 = (const float*)d_in[31];
    const float* whh0 = (const float*)d_in[32];
    const float* bih0 = (const float*)d_in[33];
    const float* bhh0 = (const float*)d_in[34];
    const float* wih1 = (const float*)d_in[35];
    const float* whh1 = (const float*)d_in[36];
    const float* bih1 = (const float*)d_in[37];
    const float* bhh1 = (const float*)d_in[38];

    // Ping-pong workspace regions (peak live ~400 MB; assumes ws_size >= ~403 MB).
    float* bufA = (float*)d_ws;
    float* bufB = (float*)((char*)d_ws + (size_t)268435456);  // +256 MiB

    // x (B,20,L) -> h0 (B,L,20) in bufA
    transpose_bcl_blc<<<2048, 256, 0, stream>>>(x, bufA, Bn, 20, L);

    const int Lin[5]  = {4096, 4076, 4056, 4036, 4030};
    const int Lout[5] = {4076, 4056, 4036, 4030, 4024};
    const int Din[5]  = {20, 256, 512, 256, 128};
    const int Dout[5] = {256, 512, 256, 128, 128};
    const int Cc[5]   = {5, 5, 5, 3, 3};
    const int Dd[5]   = {5, 5, 5, 3, 3};

    float* cur = bufA;
    float* nxt = bufB;
    for (int i = 0; i < 5; ++i) {
        int M = Bn * Lout[i];
        dim3 grid((M + MT - 1) / MT, (Dout[i] + NT - 1) / NT);
        tdnn_wmma_f32<<<grid, 256, 0, stream>>>(
            cur, W[i], Bi[i], G[i], BB[i], Mn[i], Vr[i], nxt,
            Bn, Lin[i], Lout[i], Din[i], Dout[i], Cc[i], Dd[i], /*mode=*/0);
        float* tmp = cur; cur = nxt; nxt = tmp;
    }
    // After 5 layers: activations (B,T,128) in bufB; bufA free.

    // gi0 = h @ wih0^T + bih0 : plain GEMM via same kernel (c=1, mode=1) -> bufA
    {
        int M = Bn * T;
        dim3 grid((M + MT - 1) / MT, (48 + NT - 1) / NT);
        tdnn_wmma_f32<<<grid, 256, 0, stream>>>(
            bufB, wih0, bih0, nullptr, nullptr, nullptr, nullptr, bufA,
            Bn, T, T, 128, 48, /*c=*/1, /*dil=*/0, /*mode=*/1);
    }
    // GRU0 scan: gi (bufA) -> ys0 (bufB, overwrites dead activations)
    gru_scan0<<<1, 512, 0, stream>>>(bufA, whh0, bhh0, bufB, T);
    // GRU1 scan: ys0 (bufB) -> d_out (B, T*16) fp32
    gru_scan1<<<1, 512, 0, stream>>>(bufB, wih1, whh1, bih1, bhh1,
                                     (float*)d_out, T);
}